// _MultivariatePowerFrequencyFeatures_29686813949957
// MI455X (gfx1250) — compile-verified
//
#include <hip/hip_runtime.h>
#include <math.h>

typedef float v2f __attribute__((ext_vector_type(2)));
typedef float v8f __attribute__((ext_vector_type(8)));

#define PI_F 3.14159265358979323846f

// ---------------------------------------------------------------------------
// Problem constants
// ---------------------------------------------------------------------------
#define B_SZ     8
#define C_SZ     32
#define T_SZ     16000
#define N_FFT    256
#define HOP      64
#define NFRAMES  247          // 1 + (16000-256)/64
#define NF       129          // rfft bins
#define NF_PAD   144          // padded to 9 WMMA tiles (cols 129..143 are zero)
#define WSIZE    16           // window_length/hop
#define WSTEP    4            // stride/hop
#define NWIN     58           // (247-16)/4 + 1
#define NBANDS   6
#define M_TOT    (B_SZ*C_SZ*NFRAMES)          // 63232, divisible by 16
#define SPEC_N   ((size_t)M_TOT * NF_PAD)     // 9,105,408 floats per component

static __device__ __forceinline__ v8f wmma_f32(v2f a, v2f b, v8f c) {
  return __builtin_amdgcn_wmma_f32_16x16x4_f32(false, a, false, b, (short)0, c,
                                               false, false);
}

// ---------------------------------------------------------------------------
// Kernel 1: build window-folded DFT tables  B[k][f] = w[k]*exp(-2pi i k f/256)/||w||
// dft tables are [256][144], cols 129..143 zero.  36 blocks x 256 threads,
// each block fills 1024 entries (36*1024 == 256*144).
// ---------------------------------------------------------------------------
__global__ void init_dft_kernel(float* __restrict__ dftr, float* __restrict__ dfti) {
  __shared__ float w[N_FFT];
  __shared__ float red[N_FFT];
  int tid = threadIdx.x;
  // np.hanning(256): symmetric, denominator M-1 = 255
  float wn = 0.5f - 0.5f * cosf(2.0f * PI_F * (float)tid / 255.0f);
  w[tid] = wn;
  red[tid] = wn * wn;
  __syncthreads();
  for (int s = 128; s > 0; s >>= 1) {
    if (tid < s) red[tid] += red[tid + s];
    __syncthreads();
  }
  float invn = 1.0f / sqrtf(red[0]);
  for (int j = 0; j < 4; ++j) {
    int e = blockIdx.x * 1024 + j * 256 + tid;
    int n = e / NF_PAD;
    int f = e - n * NF_PAD;
    float vr = 0.0f, vi = 0.0f;
    if (f < NF) {
      int ph = (n * f) & 255;                 // exact: exp periodic in n*f mod 256
      float ang = -2.0f * PI_F * (float)ph * (1.0f / 256.0f);
      vr = w[n] * cosf(ang) * invn;
      vi = w[n] * sinf(ang) * invn;
    }
    dftr[e] = vr;
    dfti[e] = vi;
  }
}

// ---------------------------------------------------------------------------
// Kernel 2: STFT as WMMA GEMM.
//   A[m][k] = input[bc*16000 + fr*64 + k]   (m = bc*247 + fr)
//   Bre/Bim = dft tables [k][f]
// spec stored FRAME-MAJOR:  spec_{re,im}[m * 144 + f]
//   -> no integer divides in the epilogue, lane-coalesced stores, no N bound
//      check (padded cols multiply zeroed table columns -> store zeros).
// grid (3952, 9) x 32 threads: one wave computes one 16x16 (M,F) tile.
// ---------------------------------------------------------------------------
__global__ void stft_kernel(const float* __restrict__ in,
                            const float* __restrict__ dftr,
                            const float* __restrict__ dfti,
                            float* __restrict__ sre, float* __restrict__ sim) {
  int mt = blockIdx.x;            // 0..3951
  int nt = blockIdx.y;            // 0..8
  int lane = threadIdx.x;         // 0..31
  int row = lane & 15;
  int hi  = lane >> 4;

  // A-fragment row for this lane
  int fi = mt * 16 + row;
  int bc = fi / NFRAMES;
  int fr = fi - bc * NFRAMES;
  const float* arow = in + (size_t)bc * T_SZ + (size_t)fr * HOP;
  int fcol = nt * 16 + row;       // B-fragment column (N index)

  v8f accR = {0.f,0.f,0.f,0.f,0.f,0.f,0.f,0.f};
  v8f accI = {0.f,0.f,0.f,0.f,0.f,0.f,0.f,0.f};

  for (int k0 = 0; k0 < N_FFT; k0 += 4) {
    if ((k0 & 63) == 0) {
      // stream-ahead hint: next 64-sample chunk of the frame and the
      // matching DFT-table rows (emits global_prefetch_b8; speculative,
      // non-faulting, dropped at the tail)
      __builtin_prefetch(arow + k0 + 64, 0, 1);
      __builtin_prefetch(&dftr[(k0 + 64) * NF_PAD + fcol], 0, 1);
      __builtin_prefetch(&dfti[(k0 + 64) * NF_PAD + fcol], 0, 1);
    }
    int ka = k0 + 2 * hi;         // K owned by this lane: ka, ka+1
    v2f a, br, bi;
    a.x  = arow[ka];
    a.y  = arow[ka + 1];
    br.x = dftr[ka * NF_PAD + fcol];
    br.y = dftr[(ka + 1) * NF_PAD + fcol];
    bi.x = dfti[ka * NF_PAD + fcol];
    bi.y = dfti[(ka + 1) * NF_PAD + fcol];
    accR = wmma_f32(a, br, accR);
    accI = wmma_f32(a, bi, accI);
  }

  // C/D layout: element (v, lane) = row v + 8*hi, col = lane&15
  size_t obase = ((size_t)(mt * 16) + 8 * hi) * NF_PAD + nt * 16 + row;
#pragma unroll
  for (int v = 0; v < 8; ++v) {
    size_t o = obase + (size_t)v * NF_PAD;
    sre[o] = accR[v];
    sim[o] = accI[v];
  }
}

// ---------------------------------------------------------------------------
// Kernel 3: CSD power + band averaging.
// One block (128 threads = 4 waves) per (b, t).  For each f: stage X (32x16
// complex) in LDS, each wave computes one 16x16 tile of
//   Re = Xr Xr^T + Xi Xi^T ,  Im = Xi Xr^T - Xr Xi^T   (via 4 WMMA chains)
// then accumulates (Re^2+Im^2) into the per-band LDS accumulator.
// ---------------------------------------------------------------------------
__device__ __forceinline__ int band_of(int f) {
  if (f < 1)   return -1;   // freq 0 in no band
  if (f <= 3)  return 0;    // (0,25]
  if (f <= 6)  return 1;    // (25,50]
  if (f <= 12) return 2;    // (50,100]
  if (f <= 25) return 3;    // (100,200]
  if (f <= 51) return 4;    // (200,400]
  return 5;                 // (400,1000], f<=128
}

__global__ void csd_band_kernel(const float* __restrict__ sre,
                                const float* __restrict__ sim,
                                float* __restrict__ banded) {
  int bt = blockIdx.x;            // 0..463 = b*58 + t
  int b = bt / NWIN;
  int t = bt - b * NWIN;
  int tid  = threadIdx.x;         // 0..127
  int lane = tid & 31;
  int wave = tid >> 5;            // 0..3
  int ti = wave >> 1;             // row tile of output
  int tj = wave & 1;              // col tile of output
  int row = lane & 15;
  int hi  = lane >> 4;

  __shared__ float xr[32][17];
  __shared__ float xi[32][17];
  __shared__ float bacc[NBANDS][32][32];

  float* baccf = &bacc[0][0][0];
  for (int e = tid; e < NBANDS * 1024; e += 128) baccf[e] = 0.0f;
  __syncthreads();

  for (int f = 0; f < NF; ++f) {
    // stage X[c][w] = spec[frame = (b*32+c)*247 + t*4 + w][f]  (frame-major)
    for (int e = tid; e < C_SZ * WSIZE; e += 128) {
      int c = e >> 4;
      int w = e & 15;
      size_t o = ((size_t)(b * C_SZ + c) * NFRAMES + t * WSTEP + w) * NF_PAD + f;
      xr[c][w] = sre[o];
      xi[c][w] = sim[o];
    }
    __syncthreads();

    int band = band_of(f);        // uniform across block
    if (band >= 0) {
      v8f aRR = {0.f,0.f,0.f,0.f,0.f,0.f,0.f,0.f};
      v8f aII = aRR, aIR = aRR, aRI = aRR;
      int ra = ti * 16 + row;     // A rows from tile ti
      int rb = tj * 16 + row;     // B "rows" (cols of X^T) from tile tj
#pragma unroll
      for (int k0 = 0; k0 < WSIZE; k0 += 4) {
        int k = k0 + 2 * hi;
        v2f ar, ai, br, bi;
        ar.x = xr[ra][k]; ar.y = xr[ra][k + 1];
        ai.x = xi[ra][k]; ai.y = xi[ra][k + 1];
        br.x = xr[rb][k]; br.y = xr[rb][k + 1];
        bi.x = xi[rb][k]; bi.y = xi[rb][k + 1];
        aRR = wmma_f32(ar, br, aRR);
        aII = wmma_f32(ai, bi, aII);
        aIR = wmma_f32(ai, br, aIR);
        aRI = wmma_f32(ar, bi, aRI);
      }
#pragma unroll
      for (int v = 0; v < 8; ++v) {
        float rr = aRR[v] + aII[v];      // Re(X X^H)
        float ri = aIR[v] - aRI[v];      // Im(X X^H)
        float p  = rr * rr + ri * ri;    // |.|^2 (scale 1/256 applied later)
        int c1 = ti * 16 + v + 8 * hi;
        int c2 = tj * 16 + row;
        bacc[band][c1][c2] += p;         // each element owned by one thread
      }
    }
    __syncthreads();
  }

  // banded[bt][band][c1][c2] = sum * (1/256) / count[band]
  for (int e = tid; e < NBANDS * 1024; e += 128) {
    int k  = e >> 10;
    int rc = e & 1023;
    float invc = (k == 0) ? (1.0f / 3.0f)  :
                 (k == 1) ? (1.0f / 3.0f)  :
                 (k == 2) ? (1.0f / 6.0f)  :
                 (k == 3) ? (1.0f / 13.0f) :
                 (k == 4) ? (1.0f / 26.0f) : (1.0f / 77.0f);
    banded[((size_t)bt * NBANDS + k) * 1024 + rc] =
        baccf[e] * (1.0f / 256.0f) * invc;
  }
}

// ---------------------------------------------------------------------------
// Kernel 4: symmetric eigendecomposition (cyclic Jacobi, tournament pairing)
// + matrix log reconstruction.  One wave32 per 32x32 SPD matrix.
// out[(((b*6+band)*32+i)*32+j)*58 + t] = sum_m V[i][m]*log(max(l_m,100eps))*V[j][m]
// ---------------------------------------------------------------------------
__global__ void jacobi_logm_kernel(const float* __restrict__ banded,
                                   float* __restrict__ out) {
  int blk  = blockIdx.x;          // (b*58+t)*6 + band
  int band = blk % NBANDS;
  int bt   = blk / NBANDS;
  int b    = bt / NWIN;
  int t    = bt - b * NWIN;
  int lane = threadIdx.x;         // 0..31 (single wave)

  __shared__ float A[32][33];
  __shared__ float V[32][33];
  __shared__ float cs[16], sn[16];
  __shared__ int   pp[16], qq[16];
  __shared__ float lv[32];

  const float* src = banded + (size_t)blk * 1024;
  for (int j = 0; j < 32; ++j) {
    A[j][lane] = src[j * 32 + lane];
    V[j][lane] = (j == lane) ? 1.0f : 0.0f;
  }
  __syncthreads();

  for (int sweep = 0; sweep < 12; ++sweep) {
    for (int r = 0; r < 31; ++r) {
      if (lane < 16) {
        int i0 = lane, i1 = 31 - lane;
        int p = (i0 == 0) ? 0 : 1 + (i0 - 1 + r) % 31;
        int q = 1 + (i1 - 1 + r) % 31;
        if (p > q) { int tmp = p; p = q; q = tmp; }
        float apq = A[p][q];
        float c = 1.0f, s = 0.0f;
        if (fabsf(apq) > 1e-36f) {
          float theta = (A[q][q] - A[p][p]) / (2.0f * apq);
          float tt = 1.0f / (fabsf(theta) + sqrtf(theta * theta + 1.0f));
          if (theta < 0.0f) tt = -tt;
          c = 1.0f / sqrtf(tt * tt + 1.0f);
          s = tt * c;
        }
        cs[lane] = c; sn[lane] = s; pp[lane] = p; qq[lane] = q;
      }
      __syncthreads();
      // row update (J^T A): lane = column index j
      for (int k = 0; k < 16; ++k) {
        int p = pp[k], q = qq[k];
        float c = cs[k], s = sn[k];
        float apj = A[p][lane], aqj = A[q][lane];
        A[p][lane] = c * apj - s * aqj;
        A[q][lane] = s * apj + c * aqj;
      }
      __syncthreads();
      // column update (. J) and eigenvector update: lane = row index i
      for (int k = 0; k < 16; ++k) {
        int p = pp[k], q = qq[k];
        float c = cs[k], s = sn[k];
        float aip = A[lane][p], aiq = A[lane][q];
        A[lane][p] = c * aip - s * aiq;
        A[lane][q] = s * aip + c * aiq;
        float vip = V[lane][p], viq = V[lane][q];
        V[lane][p] = c * vip - s * viq;
        V[lane][q] = s * vip + c * viq;
      }
      __syncthreads();
    }
  }

  // eigvals floored at 100*eps(f32), logged
  lv[lane] = logf(fmaxf(A[lane][lane], 100.0f * 1.1920929e-7f));
  __syncthreads();

  float wv[32];
#pragma unroll
  for (int m = 0; m < 32; ++m) wv[m] = V[lane][m] * lv[m];

  // out layout: (b, band, i, j, t); lane = row i
  float* dst = out + (((size_t)b * NBANDS + band) * 32 + lane) * 32 * NWIN + t;
  for (int j = 0; j < 32; ++j) {
    float s2 = 0.0f;
#pragma unroll
    for (int m = 0; m < 32; ++m) s2 += wv[m] * V[j][m];
    dst[(size_t)j * NWIN] = s2;
  }
}

// ---------------------------------------------------------------------------
// Host launcher
// ---------------------------------------------------------------------------
extern "C" void kernel_launch(void* const* d_in, const int* in_sizes, int n_in,
                              void* d_out, int out_size, void* d_ws, size_t ws_size,
                              hipStream_t stream) {
  (void)in_sizes; (void)n_in; (void)out_size; (void)ws_size;
  const float* x = (const float*)d_in[0];
  float* out = (float*)d_out;

  // workspace layout (floats): spec_re | spec_im | dftr | dfti | banded
  float* ws     = (float*)d_ws;
  float* sre    = ws;
  float* sim    = sre + SPEC_N;
  float* dftr   = sim + SPEC_N;
  float* dfti   = dftr + (size_t)N_FFT * NF_PAD;
  float* banded = dfti + (size_t)N_FFT * NF_PAD;

  // 1) DFT tables (window + 1/||w|| folded in)
  init_dft_kernel<<<dim3(36), dim3(256), 0, stream>>>(dftr, dfti);

  // 2) STFT via WMMA GEMM: (63232 x 256) . (256 x 144)
  stft_kernel<<<dim3(M_TOT / 16, NF_PAD / 16), dim3(32), 0, stream>>>(
      x, dftr, dfti, sre, sim);

  // 3) CSD power + band averaging: one block per (b, t)
  csd_band_kernel<<<dim3(B_SZ * NWIN), dim3(128), 0, stream>>>(sre, sim, banded);

  // 4) Jacobi eigh + matrix log: one wave per (b, t, band)
  jacobi_logm_kernel<<<dim3(B_SZ * NWIN * NBANDS), dim3(32), 0, stream>>>(
      banded, out);
}